// DenseWarpLayer_48284022342355
// MI455X (gfx1250) — compile-verified
//
#include <hip/hip_runtime.h>
#include <stdint.h>

// dense_image_warp, fp32, shapes fixed by the reference:
//   image (8,512,512,32), flow (8,512,512,2) -> out (8,512,512,32)
// Memory-bound gather: ~553 MB min traffic -> ~24us at 23.3 TB/s.

typedef float v4f __attribute__((ext_vector_type(4)));

constexpr int N = 8, H = 512, W = 512, C = 32;
constexpr int PIX_PER_BLOCK = 32;   // 8 lanes per pixel * 32 pixels = 256 threads
constexpr int THREADS = 256;

__global__ __launch_bounds__(THREADS) void DenseWarp_kernel(
    const float* __restrict__ image,
    const float* __restrict__ flow,
    float* __restrict__ out)
{
    __shared__ float sflow[PIX_PER_BLOCK * 2];

    const int t        = threadIdx.x;
    const int block_p0 = blockIdx.x * PIX_PER_BLOCK;

    // ---- Stage this block's flow tile (32 pixels x float2 = 256 B) into LDS
    // via the gfx1250 async-DMA path (ASYNCcnt-tracked, bypasses VGPRs).
    // Wave 0 (lanes 0..31) issues one B64 per pixel.
    if (t < PIX_PER_BLOCK) {
        uint32_t lds_off = (uint32_t)(uintptr_t)(&sflow[t * 2]);
        uint32_t goff    = (uint32_t)((block_p0 + t) * 2u * sizeof(float));
        asm volatile("global_load_async_to_lds_b64 %0, %1, %2"
                     :: "v"(lds_off), "v"(goff), "s"(flow) : "memory");
    }
    asm volatile("s_wait_asynccnt 0" ::: "memory");
    __syncthreads();

    const int pl = t >> 3;         // pixel within block: 0..31
    const int cq = t & 7;          // channel quad: 0..7 (4 floats each)
    const int p  = block_p0 + pl;  // global pixel index

    const int x = p & (W - 1);
    const int y = (p >> 9) & (H - 1);
    const int n = p >> 18;

    const float dy = sflow[pl * 2 + 0];   // flow[...,0] displaces y
    const float dx = sflow[pl * 2 + 1];   // flow[...,1] displaces x

    // Faithful port of tfa.dense_image_warp clamping.
    const float qy   = (float)y - dy;
    const float qx   = (float)x - dx;
    const float fy_f = fminf(fmaxf(floorf(qy), 0.0f), (float)(H - 2));
    const float fx_f = fminf(fmaxf(floorf(qx), 0.0f), (float)(W - 2));
    const float ay   = fminf(fmaxf(qy - fy_f, 0.0f), 1.0f);
    const float ax   = fminf(fmaxf(qx - fx_f, 0.0f), 1.0f);
    const int   fy   = (int)fy_f;
    const int   fx   = (int)fx_f;

    // 128-bit coalesced gathers: 8 lanes cover one pixel's 32 channels.
    const v4f* img4 = (const v4f*)image;
    const int  base = ((((n * H + fy) * W + fx) * C) >> 2) + cq;  // v4f units
    v4f tl = img4[base];
    v4f tr = img4[base + (C >> 2)];
    v4f bl = img4[base + ((W * C) >> 2)];
    v4f br = img4[base + ((W * C + C) >> 2)];

    v4f top = tl + ax * (tr - tl);
    v4f bot = bl + ax * (br - bl);
    v4f res = top + ay * (bot - top);

    // Output is write-once / never re-read: non-temporal store keeps the
    // 192 MB L2 for the image gather reuse that sets HBM traffic.
    __builtin_nontemporal_store(res, (v4f*)out + (size_t)p * (C >> 2) + cq);
}

extern "C" void kernel_launch(void* const* d_in, const int* in_sizes, int n_in,
                              void* d_out, int out_size, void* d_ws, size_t ws_size,
                              hipStream_t stream) {
    const float* image = (const float*)d_in[0];
    const float* flow  = (const float*)d_in[1];
    float*       out   = (float*)d_out;

    const int pixels = N * H * W;                       // 2,097,152
    dim3 grid(pixels / PIX_PER_BLOCK);                  // 65,536 blocks
    dim3 block(THREADS);                                // 8 waves (wave32)
    DenseWarp_kernel<<<grid, block, 0, stream>>>(image, flow, out);
}